// FLDAttention_83236466197025
// MI455X (gfx1250) — compile-verified
//
#include <hip/hip_runtime.h>

// Problem constants (match reference)
#define B_   8
#define LQ_  64
#define LK_  512
#define E_   32
#define H_   2
#define DIM_ 128
#define LAT_ 64
#define DK_  16
#define QPB  4   // query rows per softmax block (amortizes mask/value traffic)

typedef __attribute__((ext_vector_type(2))) float v2f;
typedef __attribute__((ext_vector_type(8))) float v8f;

// D = A(16x4, f32) * B(4x16, f32) + C(16x16, f32), full precision CDNA5 WMMA.
static __device__ __forceinline__ v8f wmma4(v2f a, v2f b, v8f c) {
  return __builtin_amdgcn_wmma_f32_16x16x4_f32(
      /*neg_a=*/false, a, /*neg_b=*/false, b,
      /*c_mod=*/(short)0, c, /*reuse_a=*/false, /*reuse_b=*/false);
}

// ---------------------------------------------------------------------------
// Kernel 1: q = query@Wq + bq ; k = key@Wk + bk   (rows 0..511 = q, rest = k)
// One wave per 16x16 output tile. M_total = B*(LQ+LK) = 4608, N = 32, K = 32.
// ---------------------------------------------------------------------------
__global__ __launch_bounds__(32) void proj_kernel(
    const float* __restrict__ query, const float* __restrict__ key,
    const float* __restrict__ Wq, const float* __restrict__ bq,
    const float* __restrict__ Wk, const float* __restrict__ bk,
    float* __restrict__ qbuf, float* __restrict__ kbuf) {
  const int lane = threadIdx.x & 31;
  const int half = lane >> 4;          // 0: K{0,1}/rows 0-7 ; 1: K{2,3}/rows 8-15
  const int l15  = lane & 15;
  const int tn = blockIdx.x & 1;       // E/16 = 2 tiles
  const int tm = blockIdx.x >> 1;      // 288 row tiles
  int rb = tm * 16;
  const float *X, *W, *bias; float *out;
  if (rb < B_ * LQ_) { X = query; W = Wq; bias = bq; out = qbuf; }
  else               { X = key;   W = Wk; bias = bk; out = kbuf; rb -= B_ * LQ_; }

  const int col  = tn * 16 + l15;      // N index for B/C/D lanes
  const int arow = rb + l15;           // M index for A lanes
  v8f c = {};
  for (int k0 = 0; k0 < E_; k0 += 4) {
    const int k = k0 + 2 * half;
    v2f a, b;
    a.x = X[arow * E_ + k];     a.y = X[arow * E_ + k + 1];
    b.x = W[k * E_ + col];      b.y = W[(k + 1) * E_ + col];
    c = wmma4(a, b, c);
  }
  const float bv = bias[col];
  const int r0 = rb + 8 * half;
#pragma unroll
  for (int v = 0; v < 8; ++v) out[(r0 + v) * E_ + col] = c[v] + bv;
}

// ---------------------------------------------------------------------------
// Kernel 2: scores[b,h,q,k] = (q_bh @ k_bh^T) / sqrt(DK)
// Per (b,h): M=64, N=512, K=16. One wave per 16x16 tile -> 2048 waves.
// ---------------------------------------------------------------------------
__global__ __launch_bounds__(32) void score_kernel(
    const float* __restrict__ qbuf, const float* __restrict__ kbuf,
    float* __restrict__ scores) {
  const int lane = threadIdx.x & 31;
  const int half = lane >> 4, l15 = lane & 15;
  int t = blockIdx.x;
  const int nt = t & 31; t >>= 5;      // LK/16 = 32
  const int mt = t & 3;  t >>= 2;      // LQ/16 = 4
  const int h  = t & 1;
  const int b  = t >> 1;

  const int qrow = mt * 16 + l15;
  const int kcol = nt * 16 + l15;
  v8f c = {};
  for (int k0 = 0; k0 < DK_; k0 += 4) {
    const int d = h * DK_ + k0 + 2 * half;
    v2f a, bb;
    a.x  = qbuf[(b * LQ_ + qrow) * E_ + d];
    a.y  = qbuf[(b * LQ_ + qrow) * E_ + d + 1];
    bb.x = kbuf[(b * LK_ + kcol) * E_ + d];   // B[k][n] = k_proj^T
    bb.y = kbuf[(b * LK_ + kcol) * E_ + d + 1];
    c = wmma4(a, bb, c);
  }
  float* srow = scores + (size_t)(b * H_ + h) * LQ_ * LK_;
  const int r0 = mt * 16 + 8 * half;
#pragma unroll
  for (int v = 0; v < 8; ++v)
    srow[(size_t)(r0 + v) * LK_ + kcol] = c[v] * 0.25f;   // 1/sqrt(16)
}

// ---------------------------------------------------------------------------
// Kernel 3: per-channel masked softmax over Lk, then weighted sum with value.
// Block = 128 threads (one channel each), handles QPB query rows of one (b,h).
// Score rows staged in LDS; two passes (max, then exp/accumulate) reproduce
// reference semantics exactly, including the all-masked uniform edge case.
// ---------------------------------------------------------------------------
__global__ __launch_bounds__(128) void softmax_av_kernel(
    const float* __restrict__ scores, const int* __restrict__ mask,
    const float* __restrict__ value, float* __restrict__ xbuf) {
  __shared__ float s_lds[QPB][LK_];
  int blk = blockIdx.x;
  const int qg = blk & (LQ_ / QPB - 1); blk >>= 4;   // LQ/QPB = 16
  const int h  = blk & 1;
  const int b  = blk >> 1;
  const int q0 = qg * QPB;
  const int tid = threadIdx.x;

  const float* srow = scores + ((size_t)(b * H_ + h) * LQ_ + q0) * LK_;
  for (int i = tid; i < QPB * LK_; i += 128)
    s_lds[i >> 9][i & (LK_ - 1)] = srow[i];          // rows are contiguous
  __syncthreads();

  const int c = tid;                                 // channel 0..127
  const int*   mrow = mask  + (size_t)b * LK_ * DIM_ + c;
  const float* vrow = value + (size_t)b * LK_ * DIM_ + c;

  float m[QPB];
#pragma unroll
  for (int qi = 0; qi < QPB; ++qi) m[qi] = -3.0e38f;

  for (int k = 0; k < LK_; ++k) {                    // pass 1: channel-wise max
    const int mk = mrow[k * DIM_];
#pragma unroll
    for (int qi = 0; qi < QPB; ++qi) {
      const float se = mk ? s_lds[qi][k] : -1.0e9f;
      m[qi] = fmaxf(m[qi], se);
    }
  }

  float Z[QPB], num[QPB];
#pragma unroll
  for (int qi = 0; qi < QPB; ++qi) { Z[qi] = 0.f; num[qi] = 0.f; }

  for (int k = 0; k < LK_; ++k) {                    // pass 2: exp + weighted sum
    const int   mk  = mrow[k * DIM_];
    const float val = vrow[k * DIM_];
#pragma unroll
    for (int qi = 0; qi < QPB; ++qi) {
      const float se = mk ? s_lds[qi][k] : -1.0e9f;
      const float e  = __expf(se - m[qi]);
      Z[qi]   += e;
      num[qi] += e * val;
    }
  }

#pragma unroll
  for (int qi = 0; qi < QPB; ++qi) {
    const int q = q0 + qi;                           // store as [b, q, h*DIM + c]
    xbuf[((size_t)(b * LQ_ + q)) * (H_ * DIM_) + h * DIM_ + c] = num[qi] / Z[qi];
  }
}

// ---------------------------------------------------------------------------
// Kernel 4: out = x[512,256] @ Wout[256,64] + bout. One wave per 16x16 tile.
// ---------------------------------------------------------------------------
__global__ __launch_bounds__(32) void out_kernel(
    const float* __restrict__ xbuf, const float* __restrict__ Wout,
    const float* __restrict__ bout, float* __restrict__ out) {
  const int lane = threadIdx.x & 31;
  const int half = lane >> 4, l15 = lane & 15;
  const int tn = blockIdx.x & 3;       // LAT/16 = 4
  const int tm = blockIdx.x >> 2;      // (B*LQ)/16 = 32
  const int row = tm * 16 + l15;
  const int col = tn * 16 + l15;
  const int KD = H_ * DIM_;            // 256
  v8f c = {};
  for (int k0 = 0; k0 < KD; k0 += 4) {
    const int k = k0 + 2 * half;
    v2f a, b;
    a.x = xbuf[row * KD + k];      a.y = xbuf[row * KD + k + 1];
    b.x = Wout[k * LAT_ + col];    b.y = Wout[(k + 1) * LAT_ + col];
    c = wmma4(a, b, c);
  }
  const float bv = bout[col];
  const int r0 = tm * 16 + 8 * half;
#pragma unroll
  for (int v = 0; v < 8; ++v) out[(r0 + v) * LAT_ + col] = c[v] + bv;
}

// ---------------------------------------------------------------------------
extern "C" void kernel_launch(void* const* d_in, const int* in_sizes, int n_in,
                              void* d_out, int out_size, void* d_ws, size_t ws_size,
                              hipStream_t stream) {
  (void)in_sizes; (void)n_in; (void)out_size; (void)ws_size;
  const float* query = (const float*)d_in[0];
  const float* key   = (const float*)d_in[1];
  const float* value = (const float*)d_in[2];
  const int*   mask  = (const int*)  d_in[3];
  const float* Wq    = (const float*)d_in[4];
  const float* bq    = (const float*)d_in[5];
  const float* Wk    = (const float*)d_in[6];
  const float* bk    = (const float*)d_in[7];
  const float* Wout  = (const float*)d_in[8];
  const float* bout  = (const float*)d_in[9];
  float* out = (float*)d_out;

  // Workspace layout (floats): qbuf 16K | kbuf 128K | scores 512K | xbuf 128K
  float* ws     = (float*)d_ws;
  float* qbuf   = ws;                                   // [B*LQ, E]
  float* kbuf   = qbuf + B_ * LQ_ * E_;                 // [B*LK, E]
  float* scores = kbuf + B_ * LK_ * E_;                 // [B,H,LQ,LK]
  float* xbuf   = scores + (size_t)B_ * H_ * LQ_ * LK_; // [B*LQ, H*DIM]

  proj_kernel<<<(B_ * (LQ_ + LK_) / 16) * (E_ / 16), 32, 0, stream>>>(
      query, key, Wq, bq, Wk, bk, qbuf, kbuf);
  score_kernel<<<B_ * H_ * (LQ_ / 16) * (LK_ / 16), 32, 0, stream>>>(
      qbuf, kbuf, scores);
  softmax_av_kernel<<<B_ * H_ * (LQ_ / QPB), 128, 0, stream>>>(
      scores, mask, value, xbuf);
  out_kernel<<<(B_ * LQ_ / 16) * (LAT_ / 16), 32, 0, stream>>>(
      xbuf, Wout, bout, out);
}